// GCN_16295105921346
// MI455X (gfx1250) — compile-verified
//
#include <hip/hip_runtime.h>
#include <stdint.h>

// ---------------------------------------------------------------------------
// GCN forward: 2x (gather-scale, segment_max, tiny GEMM) on MI455X (gfx1250).
// segment_max via order-preserving u32 encoding + native GLOBAL_ATOMIC_MAX_U32
// (all node state fits in 192MB L2). GEMM epilogues via V_WMMA_F32_16X16X4_F32
// (one 16-node tile per wave32, full f32 precision).
// ---------------------------------------------------------------------------

typedef float v2f __attribute__((ext_vector_type(2)));
typedef float v8f __attribute__((ext_vector_type(8)));

#define ORD_NEG_INF 0x007FFFFFu   // order-key of -inf (minimum attainable key)

__device__ __forceinline__ unsigned ord_encode(float f) {
    unsigned u = __float_as_uint(f);
    return (u & 0x80000000u) ? ~u : (u | 0x80000000u);
}

// decode order-key; non-finite (i.e. untouched -inf slots) -> 0.0f, matching
// the reference's jnp.where(isfinite(agg), agg, 0).
__device__ __forceinline__ float ord_decode_finite(unsigned k) {
    unsigned u = (k & 0x80000000u) ? (k & 0x7FFFFFFFu) : ~k;
    if ((u & 0x7F800000u) == 0x7F800000u) return 0.0f;
    return __uint_as_float(u);
}

// ---------------------------------------------------------------------------
__global__ void GCN_fill_u32(unsigned* __restrict__ p, long n, unsigned v) {
    long i = blockIdx.x * (long)blockDim.x + threadIdx.x;
    if (i < n) p[i] = v;
}

// One lane per (edge, feature). Lanes in a 2^LOG2F group share an edge, so
// src/dst/ts loads broadcast and the feature gather + atomic are coalesced.
template <int LOG2F>
__global__ void GCN_edge_max(const float* __restrict__ feat,
                             const int* __restrict__ src,
                             const int* __restrict__ dst,
                             const float* __restrict__ ts,
                             unsigned* __restrict__ agg,
                             long nwork) {
    long t = blockIdx.x * (long)blockDim.x + threadIdx.x;
    if (t >= nwork) return;
    const int F = 1 << LOG2F;
    int e = (int)(t >> LOG2F);
    int f = (int)(t & (F - 1));
    int s = src[e];
    int d = dst[e];
    float m = feat[(long)s * F + f] * ts[e];
    atomicMax(agg + (long)d * F + f, ord_encode(m));   // global_atomic_max_u32
}

// ---------------------------------------------------------------------------
// Layer-1 epilogue: per wave32, one 16-node tile.
// D(16x16) = agg(16x16) @ W1pad(16x16) via 4 chained V_WMMA_F32_16X16X4_F32.
// Columns 0..7 of D are h = relu(agg@W1 + b1).
__global__ void __launch_bounds__(256)
GCN_finalize1(const unsigned* __restrict__ agg,
              const float* __restrict__ W1,   // [16][8] row-major
              const float* __restrict__ b1,   // [8]
              float* __restrict__ h,          // [N][8]
              int n_nodes) {
    int wave = (int)((blockIdx.x * (long)blockDim.x + threadIdx.x) >> 5);
    int lane = threadIdx.x & 31;
    int base = wave * 16;                 // tile base node (wave-uniform)
    if (base >= n_nodes) return;          // uniform exit: EXEC stays all-1s

    int m  = lane & 15;                   // A row / B,C,D column for this lane
    int kh = lane >> 4;                   // K-half select (ISA A/B f32 layout)
    int n  = m;

    v8f acc = {0.f, 0.f, 0.f, 0.f, 0.f, 0.f, 0.f, 0.f};
#pragma unroll
    for (int c = 0; c < 4; ++c) {
        int k0 = 4 * c + 2 * kh;
        v2f a, b;
        a.x = ord_decode_finite(agg[(base + m) * 16 + k0]);
        a.y = ord_decode_finite(agg[(base + m) * 16 + k0 + 1]);
        b.x = (n < 8) ? W1[k0 * 8 + n]       : 0.0f;   // W1 zero-padded to 4x16
        b.y = (n < 8) ? W1[(k0 + 1) * 8 + n] : 0.0f;
        acc = __builtin_amdgcn_wmma_f32_16x16x4_f32(
            /*neg_a=*/false, a, /*neg_b=*/false, b,
            /*c_mod=*/(short)0, acc, /*reuse_a=*/false, /*reuse_b=*/false);
    }

    if (n < 8) {                          // valid output columns only
#pragma unroll
        for (int r = 0; r < 8; ++r) {
            int node = base + r + 8 * kh; // D layout: vgpr r -> M = r + 8*half
            float v = acc[r] + b1[n];
            h[node * 8 + n] = v > 0.0f ? v : 0.0f;
        }
    }
}

// Layer-2 epilogue: D(16x16) = agg2(16x8) @ W2pad(8x16), 2 chained WMMAs.
// Columns 0..1 of D are the classifier output.
__global__ void __launch_bounds__(256)
GCN_finalize2(const unsigned* __restrict__ agg,
              const float* __restrict__ W2,   // [8][2] row-major
              const float* __restrict__ b2,   // [2]
              float* __restrict__ out,        // [N][2]
              int n_nodes) {
    int wave = (int)((blockIdx.x * (long)blockDim.x + threadIdx.x) >> 5);
    int lane = threadIdx.x & 31;
    int base = wave * 16;
    if (base >= n_nodes) return;

    int m  = lane & 15;
    int kh = lane >> 4;
    int n  = m;

    v8f acc = {0.f, 0.f, 0.f, 0.f, 0.f, 0.f, 0.f, 0.f};
#pragma unroll
    for (int c = 0; c < 2; ++c) {
        int k0 = 4 * c + 2 * kh;
        v2f a, b;
        a.x = ord_decode_finite(agg[(base + m) * 8 + k0]);
        a.y = ord_decode_finite(agg[(base + m) * 8 + k0 + 1]);
        b.x = (n < 2) ? W2[k0 * 2 + n]       : 0.0f;   // W2 zero-padded to 4x16
        b.y = (n < 2) ? W2[(k0 + 1) * 2 + n] : 0.0f;
        acc = __builtin_amdgcn_wmma_f32_16x16x4_f32(
            false, a, false, b, (short)0, acc, false, false);
    }

    if (n < 2) {
#pragma unroll
        for (int r = 0; r < 8; ++r) {
            int node = base + r + 8 * kh;
            out[node * 2 + n] = acc[r] + b2[n];
        }
    }
}

// ---------------------------------------------------------------------------
extern "C" void kernel_launch(void* const* d_in, const int* in_sizes, int n_in,
                              void* d_out, int out_size, void* d_ws, size_t ws_size,
                              hipStream_t stream) {
    const float* x   = (const float*)d_in[0];
    const int*   src = (const int*)  d_in[1];
    const int*   dst = (const int*)  d_in[2];
    const float* ts  = (const float*)d_in[3];
    const float* W1  = (const float*)d_in[4];
    const float* b1  = (const float*)d_in[5];
    const float* W2  = (const float*)d_in[6];
    const float* b2  = (const float*)d_in[7];
    float* out = (float*)d_out;

    const int N = in_sizes[0] / 16;   // 100000 nodes (IN_FEATS=16)
    const int E = in_sizes[1];        // 6.5M edges

    // Workspace: agg1[N*16] u32 | h[N*8] f32 | agg2[N*8] u32  (= N*128 bytes)
    unsigned* agg1 = (unsigned*)d_ws;
    float*    h    = (float*)(agg1 + (size_t)N * 16);
    unsigned* agg2 = (unsigned*)(h + (size_t)N * 8);

    const int B = 256;

    // ---- layer 1 ----
    long n1 = (long)N * 16;
    GCN_fill_u32<<<(unsigned)((n1 + B - 1) / B), B, 0, stream>>>(agg1, n1, ORD_NEG_INF);

    long w1 = (long)E * 16;
    GCN_edge_max<4><<<(unsigned)((w1 + B - 1) / B), B, 0, stream>>>(x, src, dst, ts, agg1, w1);

    int waves   = N / 16;                       // 6250 tiles, one per wave32
    int fin_thr = waves * 32;
    GCN_finalize1<<<(unsigned)((fin_thr + B - 1) / B), B, 0, stream>>>(agg1, W1, b1, h, N);

    // ---- layer 2 ----
    long n2 = (long)N * 8;
    GCN_fill_u32<<<(unsigned)((n2 + B - 1) / B), B, 0, stream>>>(agg2, n2, ORD_NEG_INF);

    long w2 = (long)E * 8;
    GCN_edge_max<3><<<(unsigned)((w2 + B - 1) / B), B, 0, stream>>>(h, src, dst, ts, agg2, w2);

    GCN_finalize2<<<(unsigned)((fin_thr + B - 1) / B), B, 0, stream>>>(agg2, W2, b2, out, N);
}